// HeteroGAT_32495722561819
// MI455X (gfx1250) — compile-verified
//
#include <hip/hip_runtime.h>
#include <hip/hip_bf16.h>

typedef __attribute__((ext_vector_type(2))) float v2f;
typedef __attribute__((ext_vector_type(4))) float v4f;
typedef __attribute__((ext_vector_type(8))) float v8f;

#define NEG_SLOPE 0.2f
#define SOFTMAX_EPS 1e-16f

// ---------------------------------------------------------------------------
// Generic fill (grid-stride)
// ---------------------------------------------------------------------------
__global__ void hg_fill(float* __restrict__ p, float v, long n) {
    long i = (long)blockIdx.x * blockDim.x + threadIdx.x;
    long stride = (long)gridDim.x * blockDim.x;
    for (; i < n; i += stride) p[i] = v;
}

// ---------------------------------------------------------------------------
// FP32 WMMA GEMM:  C[N,128] = X[N,K] @ W[K,128] (+ optional bias[128])
//
// Block = 256 threads (8 wave32) computes a 64x128 output tile:
//  - The 64xK A panel is cooperatively staged into LDS (float4 global loads,
//    row stride K+4 words so the ds_load_b64 A-fragment reads are
//    bank-conflict-free for K in {32,64,128}).
//  - Wave w owns output columns [16w, 16w+16). Per k-step it loads ONE B
//    fragment from global (L2-resident W) and reuses it across 4 row tiles:
//    4 x v_wmma_f32_16x16x4_f32 per B fragment.
// Requires N % 16 == 0, K % 4 == 0 (true for all call sites).
//
// WMMA fp32 lane layout:
//  A (16x4): lanes 0-15 -> M=lane, {K=k0,k0+1}; lanes 16-31 -> {K=k0+2,k0+3}
//  B (4x16): VGPR0 = K=k0 (lo half) / k0+2 (hi half); VGPR1 = k0+1 / k0+3
//  C/D: VGPR i -> M = i + 8*(lane/16), N = lane%16
// ---------------------------------------------------------------------------
#define GEMM_ROWS 64
#define LDS_STRIDE_PAD 4

__global__ __launch_bounds__(256)
void hg_gemm_wmma_f32(const float* __restrict__ X, const float* __restrict__ W,
                      const float* __restrict__ bias, float* __restrict__ C,
                      int N, int K) {
    __shared__ float As[GEMM_ROWS * (128 + LDS_STRIDE_PAD)];
    const int strideA = K + LDS_STRIDE_PAD;
    const int tid  = threadIdx.x;
    const int wave = tid >> 5;
    const int lane = tid & 31;
    const int hl   = lane >> 4;
    const int r    = lane & 15;
    const int m0   = blockIdx.x * GEMM_ROWS;
    const int n0   = wave * 16;

    // --- cooperative stage of A panel (64 x K) into LDS, zero-fill OOB rows
    const int totalA = GEMM_ROWS * K;
    for (int i = tid * 4; i < totalA; i += 256 * 4) {
        const int row = i / K;
        const int col = i - row * K;
        v4f val = {0.f, 0.f, 0.f, 0.f};
        if (m0 + row < N)
            val = *(const v4f*)(X + (size_t)(m0 + row) * K + col);
        float* dst = As + row * strideA + col;
        dst[0] = val.x; dst[1] = val.y; dst[2] = val.z; dst[3] = val.w;
    }
    __syncthreads();

    v8f acc0 = {0.f,0.f,0.f,0.f,0.f,0.f,0.f,0.f};
    v8f acc1 = acc0, acc2 = acc0, acc3 = acc0;

    for (int k0 = 0; k0 < K; k0 += 4) {
        const int ka = k0 + 2 * hl;
        v2f b;
        b.x = W[(size_t)ka * 128 + n0 + r];
        b.y = W[(size_t)(ka + 1) * 128 + n0 + r];
        v2f a0, a1, a2, a3;
        const float* Ab = As + r * strideA + ka;
        a0.x = Ab[0];                a0.y = Ab[1];
        a1.x = Ab[16 * strideA];     a1.y = Ab[16 * strideA + 1];
        a2.x = Ab[32 * strideA];     a2.y = Ab[32 * strideA + 1];
        a3.x = Ab[48 * strideA];     a3.y = Ab[48 * strideA + 1];
        acc0 = __builtin_amdgcn_wmma_f32_16x16x4_f32(false, a0, false, b, (short)0, acc0, false, false);
        acc1 = __builtin_amdgcn_wmma_f32_16x16x4_f32(false, a1, false, b, (short)0, acc1, false, false);
        acc2 = __builtin_amdgcn_wmma_f32_16x16x4_f32(false, a2, false, b, (short)0, acc2, false, false);
        acc3 = __builtin_amdgcn_wmma_f32_16x16x4_f32(false, a3, false, b, (short)0, acc3, false, false);
    }

    const float bb = bias ? bias[n0 + r] : 0.f;
    v8f accs[4] = {acc0, acc1, acc2, acc3};
#pragma unroll
    for (int rt = 0; rt < 4; ++rt) {
        const int mt = m0 + rt * 16;
        if (mt >= N) break;             // block-uniform: EXEC stays full
#pragma unroll
        for (int i = 0; i < 8; ++i) {
            const int row = mt + i + 8 * hl;
            C[(size_t)row * 128 + n0 + r] = accs[rt][i] + bb;
        }
    }
}

// ---------------------------------------------------------------------------
// a_src[n,h] = sum_c hs[n, h*32+c] * att_src[h*32+c]
// ---------------------------------------------------------------------------
__global__ void hg_attn_src(const float* __restrict__ hs,
                            const float* __restrict__ att,
                            float* __restrict__ a, int N4) {
    int i = blockIdx.x * blockDim.x + threadIdx.x;
    if (i >= N4) return;
    const int n = i >> 2, h = i & 3;
    const float* row = hs + (size_t)n * 128 + h * 32;
    const float* av  = att + h * 32;
    float s = 0.f;
#pragma unroll
    for (int c = 0; c < 32; ++c) s += row[c] * av[c];
    a[i] = s;
}

// v[k*4+h] = sum_c W_dst[k*128 + h*32 + c] * att_dst[h*32+c]   (512 outputs)
__global__ void hg_vdst(const float* __restrict__ Wd,
                        const float* __restrict__ att, float* __restrict__ v) {
    int i = blockIdx.x * blockDim.x + threadIdx.x;
    if (i >= 512) return;
    const int k = i >> 2, h = i & 3;
    float s = 0.f;
#pragma unroll
    for (int c = 0; c < 32; ++c) s += Wd[(size_t)k * 128 + h * 32 + c] * att[h * 32 + c];
    v[i] = s;
}

// a_dst[n,h] = sum_k x_dst[n,k] * v[k*4+h]
__global__ void hg_adst(const float* __restrict__ x, const float* __restrict__ v,
                        float* __restrict__ a, int N4) {
    int i = blockIdx.x * blockDim.x + threadIdx.x;
    if (i >= N4) return;
    const int n = i >> 2, h = i & 3;
    const float* row = x + (size_t)n * 128;
    float s = 0.f;
#pragma unroll 8
    for (int k = 0; k < 128; ++k) s += row[k] * v[k * 4 + h];
    a[i] = s;
}

// ---------------------------------------------------------------------------
// Segment softmax over edges (atomics): pass1 max, pass2 exp + denom
// ---------------------------------------------------------------------------
__device__ inline void hg_atomic_max_f32(float* addr, float v) {
    if (v >= 0.f) atomicMax((int*)addr, __float_as_int(v));
    else          atomicMin((unsigned int*)addr, __float_as_uint(v));
}

__device__ inline float hg_leaky(float v) {
    return v > 0.f ? v : v * NEG_SLOPE;
}

__global__ void hg_edge_max(const int* __restrict__ src, const int* __restrict__ dst,
                            const float* __restrict__ a_src, const float* __restrict__ a_dst,
                            float* __restrict__ m, int E4) {
    int i = blockIdx.x * blockDim.x + threadIdx.x;
    if (i >= E4) return;
    const int e = i >> 2, h = i & 3;
    const int s = src[e], d = dst[e];
    const float v = hg_leaky(a_src[s * 4 + h] + a_dst[d * 4 + h]);
    hg_atomic_max_f32(&m[d * 4 + h], v);
}

__global__ void hg_edge_exp(const int* __restrict__ src, const int* __restrict__ dst,
                            const float* __restrict__ a_src, const float* __restrict__ a_dst,
                            const float* __restrict__ m, float* __restrict__ ex_buf,
                            float* __restrict__ den, int E4) {
    int i = blockIdx.x * blockDim.x + threadIdx.x;
    if (i >= E4) return;
    const int e = i >> 2, h = i & 3;
    const int s = src[e], d = dst[e];
    const float v  = hg_leaky(a_src[s * 4 + h] + a_dst[d * 4 + h]);
    const float ex = expf(v - m[d * 4 + h]);
    ex_buf[i] = ex;
    atomicAdd(&den[d * 4 + h], ex);
}

// scatter: out[dst,f] += hs[src,f] * ex[e,h] / (den[dst,h] + eps), f = 0..127
__global__ void hg_scatter(const int* __restrict__ src, const int* __restrict__ dst,
                           const float* __restrict__ hs, const float* __restrict__ ex_buf,
                           const float* __restrict__ den, float* __restrict__ out,
                           long total) {
    long i = (long)blockIdx.x * blockDim.x + threadIdx.x;
    if (i >= total) return;
    const int e = (int)(i >> 7);
    const int f = (int)(i & 127);
    const int h = f >> 5;
    const int s = src[e], d = dst[e];
    const float alpha = ex_buf[e * 4 + h] / (den[d * 4 + h] + SOFTMAX_EPS);
    atomicAdd(&out[(size_t)d * 128 + f], hs[(size_t)s * 128 + f] * alpha);
}

// ---------------------------------------------------------------------------
// Combine / bias / relu (in place)
// ---------------------------------------------------------------------------
__global__ void hg_finalize_dual(float* __restrict__ acc, const float* __restrict__ b1,
                                 const float* __restrict__ b2, long n128, int relu) {
    long i = (long)blockIdx.x * blockDim.x + threadIdx.x;
    if (i >= n128) return;
    const int f = (int)(i & 127);
    float v = 0.5f * acc[i] + 0.5f * (b1[f] + b2[f]);
    if (relu) v = fmaxf(v, 0.f);
    acc[i] = v;
}

__global__ void hg_finalize_single(float* __restrict__ acc, const float* __restrict__ b,
                                   long n128, int relu) {
    long i = (long)blockIdx.x * blockDim.x + threadIdx.x;
    if (i >= n128) return;
    const int f = (int)(i & 127);
    float v = acc[i] + b[f];
    if (relu) v = fmaxf(v, 0.f);
    acc[i] = v;
}

// ---------------------------------------------------------------------------
// Mean pooling + head
// ---------------------------------------------------------------------------
__global__ void hg_pool(const float* __restrict__ h, const int* __restrict__ batch,
                        float* __restrict__ pooled, float* __restrict__ cnt, long n128) {
    long i = (long)blockIdx.x * blockDim.x + threadIdx.x;
    if (i >= n128) return;
    const int n = (int)(i >> 7);
    const int f = (int)(i & 127);
    const int g = batch[n];
    atomicAdd(&pooled[(size_t)g * 128 + f], h[i]);
    if (f == 0) atomicAdd(&cnt[g], 1.0f);
}

__global__ __launch_bounds__(128)
void hg_head(const float* __restrict__ p0, const float* __restrict__ c0,
             const float* __restrict__ p1, const float* __restrict__ c1,
             const float* __restrict__ p2, const float* __restrict__ c2,
             const float* __restrict__ Wl, const float* __restrict__ bl,
             float* __restrict__ out) {
    __shared__ float red[128][2];
    const int g = blockIdx.x;
    const int f = threadIdx.x;
    const float e0 = p0[(size_t)g * 128 + f] / fmaxf(c0[g], 1.f);
    const float e1 = p1[(size_t)g * 128 + f] / fmaxf(c1[g], 1.f);
    const float e2 = p2[(size_t)g * 128 + f] / fmaxf(c2[g], 1.f);
    const float emb = (e0 + e1 + e2) * (1.f / 3.f);
    red[f][0] = emb * Wl[f * 2 + 0];
    red[f][1] = emb * Wl[f * 2 + 1];
    __syncthreads();
    for (int s = 64; s > 0; s >>= 1) {
        if (f < s) { red[f][0] += red[f + s][0]; red[f][1] += red[f + s][1]; }
        __syncthreads();
    }
    if (f < 2) out[g * 2 + f] = red[0][f] + bl[f];
}

// ---------------------------------------------------------------------------
// Host orchestration
// ---------------------------------------------------------------------------
extern "C" void kernel_launch(void* const* d_in, const int* in_sizes, int n_in,
                              void* d_out, int out_size, void* d_ws, size_t ws_size,
                              hipStream_t stream) {
    (void)n_in; (void)ws_size;
    const float* x_central = (const float*)d_in[0];
    const float* x_cc_in   = (const float*)d_in[1];
    const float* x_cg_in   = (const float*)d_in[2];
    const int* ei_c2cc = (const int*)d_in[3];
    const int* ei_cc2c = (const int*)d_in[4];
    const int* ei_c2cg = (const int*)d_in[5];
    const int* ei_cg2c = (const int*)d_in[6];
    const int* batch_c  = (const int*)d_in[7];
    const int* batch_cc = (const int*)d_in[8];
    const int* batch_cg = (const int*)d_in[9];

    const int N_C  = in_sizes[0] / 64;
    const int N_CC = in_sizes[1] / 32;
    const int N_CG = in_sizes[2] / 32;
    const int E    = in_sizes[3] / 2;
    const int G    = out_size / 2;

    // params flattened: 'gat' (2 layers x {c2cc,c2cg,cc2c,cg2c} x
    // {W_dst,W_src,att_dst,att_src,b}), then 'in_proj'
    // {central, child_categ, child_cont} x {W,b}, then 'lin' {W,b}
    auto P = [&](int i) { return (const float*)d_in[i]; };
    auto gbase = [&](int layer, int t) { return 10 + layer * 20 + t * 5; };
    const int T_C2CC = 0, T_C2CG = 1, T_CC2C = 2, T_CG2C = 3;
    const int IP_CENTRAL_W = 50, IP_CENTRAL_B = 51;
    const int IP_CATEG_W = 52, IP_CATEG_B = 53;
    const int IP_CONT_W = 54, IP_CONT_B = 55;
    const int LIN_W = 56, LIN_B = 57;

    // ---- workspace carve-up (floats) ----
    float* ws = (float*)d_ws;
    size_t off = 0;
    auto alloc = [&](size_t n) { float* p = ws + off; off += n; return p; };
    const size_t maxN = (size_t)((N_CC > N_CG ? N_CC : N_CG) > N_C
                                     ? (N_CC > N_CG ? N_CC : N_CG) : N_C);

    float* xC  = alloc((size_t)N_C * 128);
    float* xCC = alloc((size_t)N_CC * 128);
    float* xCG = alloc((size_t)N_CG * 128);
    float* yC  = alloc((size_t)N_C * 128);
    float* yCC = alloc((size_t)N_CC * 128);
    float* yCG = alloc((size_t)N_CG * 128);
    float* hs    = alloc(maxN * 128);
    float* a_src = alloc(maxN * 4);
    float* a_dst = alloc(maxN * 4);
    float* mbuf  = alloc(maxN * 4);
    float* den   = alloc(maxN * 4);
    float* exb   = alloc((size_t)E * 4);
    float* vdst  = alloc(512);
    float* pooled0 = alloc((size_t)G * 128);
    float* pooled1 = alloc((size_t)G * 128);
    float* pooled2 = alloc((size_t)G * 128);
    float* cnt0 = alloc(G);
    float* cnt1 = alloc(G);
    float* cnt2 = alloc(G);

    auto fill = [&](float* p, float v, long n) {
        hg_fill<<<4096, 256, 0, stream>>>(p, v, n);
    };
    auto gemm = [&](const float* X, const float* W, const float* bias, float* C,
                    int N, int K) {
        hg_gemm_wmma_f32<<<(N + GEMM_ROWS - 1) / GEMM_ROWS, 256, 0, stream>>>(
            X, W, bias, C, N, K);
    };

    // ---- input projections (WMMA) ----
    gemm(x_central, P(IP_CENTRAL_W), P(IP_CENTRAL_B), xC, N_C, 64);
    gemm(x_cc_in, P(IP_CONT_W), P(IP_CONT_B), xCC, N_CC, 32);
    gemm(x_cg_in, P(IP_CATEG_W), P(IP_CATEG_B), xCG, N_CG, 32);

    // ---- one bipartite GATConv, accumulating segment-sum into `accum` ----
    auto run_conv = [&](const float* xs, int Ns, const float* xd, int Nd,
                        const int* ei, int pbase, float* accum) {
        const int* src = ei;
        const int* dst = ei + E;
        const float* Wdst = P(pbase + 0);
        const float* Wsrc = P(pbase + 1);
        const float* attD = P(pbase + 2);
        const float* attS = P(pbase + 3);

        gemm(xs, Wsrc, nullptr, hs, Ns, 128);
        hg_attn_src<<<(Ns * 4 + 255) / 256, 256, 0, stream>>>(hs, attS, a_src, Ns * 4);
        hg_vdst<<<2, 256, 0, stream>>>(Wdst, attD, vdst);
        hg_adst<<<(Nd * 4 + 255) / 256, 256, 0, stream>>>(xd, vdst, a_dst, Nd * 4);

        fill(mbuf, -3.402823466e38f, (long)Nd * 4);
        fill(den, 0.f, (long)Nd * 4);
        hg_edge_max<<<(E * 4 + 255) / 256, 256, 0, stream>>>(src, dst, a_src, a_dst, mbuf, E * 4);
        hg_edge_exp<<<(E * 4 + 255) / 256, 256, 0, stream>>>(src, dst, a_src, a_dst, mbuf, exb, den, E * 4);
        const long total = (long)E * 128;
        hg_scatter<<<(unsigned)((total + 255) / 256), 256, 0, stream>>>(
            src, dst, hs, exb, den, accum, total);
    };

    // ---- 2 GAT layers ----
    float *cC = xC, *cCC = xCC, *cCG = xCG;   // current
    float *nC = yC, *nCC = yCC, *nCG = yCG;   // next
    for (int li = 0; li < 2; ++li) {
        const int relu = (li == 0) ? 1 : 0;

        // central <- mean(cc2c, cg2c)
        fill(nC, 0.f, (long)N_C * 128);
        run_conv(cCC, N_CC, cC, N_C, ei_cc2c, gbase(li, T_CC2C), nC);
        run_conv(cCG, N_CG, cC, N_C, ei_cg2c, gbase(li, T_CG2C), nC);
        hg_finalize_dual<<<(unsigned)(((long)N_C * 128 + 255) / 256), 256, 0, stream>>>(
            nC, P(gbase(li, T_CC2C) + 4), P(gbase(li, T_CG2C) + 4), (long)N_C * 128, relu);

        // child_cont <- c2cc
        fill(nCC, 0.f, (long)N_CC * 128);
        run_conv(cC, N_C, cCC, N_CC, ei_c2cc, gbase(li, T_C2CC), nCC);
        hg_finalize_single<<<(unsigned)(((long)N_CC * 128 + 255) / 256), 256, 0, stream>>>(
            nCC, P(gbase(li, T_C2CC) + 4), (long)N_CC * 128, relu);

        // child_categ <- c2cg
        fill(nCG, 0.f, (long)N_CG * 128);
        run_conv(cC, N_C, cCG, N_CG, ei_c2cg, gbase(li, T_C2CG), nCG);
        hg_finalize_single<<<(unsigned)(((long)N_CG * 128 + 255) / 256), 256, 0, stream>>>(
            nCG, P(gbase(li, T_C2CG) + 4), (long)N_CG * 128, relu);

        // swap current <-> next
        float* t;
        t = cC;  cC  = nC;  nC  = t;
        t = cCC; cCC = nCC; nCC = t;
        t = cCG; cCG = nCG; nCG = t;
    }

    // ---- mean pooling per node type ----
    fill(pooled0, 0.f, (long)G * 128);
    fill(pooled1, 0.f, (long)G * 128);
    fill(pooled2, 0.f, (long)G * 128);
    fill(cnt0, 0.f, G);
    fill(cnt1, 0.f, G);
    fill(cnt2, 0.f, G);
    hg_pool<<<(unsigned)(((long)N_C * 128 + 255) / 256), 256, 0, stream>>>(
        cC, batch_c, pooled0, cnt0, (long)N_C * 128);
    hg_pool<<<(unsigned)(((long)N_CC * 128 + 255) / 256), 256, 0, stream>>>(
        cCC, batch_cc, pooled1, cnt1, (long)N_CC * 128);
    hg_pool<<<(unsigned)(((long)N_CG * 128 + 255) / 256), 256, 0, stream>>>(
        cCG, batch_cg, pooled2, cnt2, (long)N_CG * 128);

    // ---- head: stack-mean + linear [128 -> 2] ----
    hg_head<<<G, 128, 0, stream>>>(pooled0, cnt0, pooled1, cnt1, pooled2, cnt2,
                                   P(LIN_W), P(LIN_B), (float*)d_out);
}